// ContextualAttention_44933947850878
// MI455X (gfx1250) — compile-verified
//
#include <hip/hip_runtime.h>

// ---------------------------------------------------------------------------
// ContextualAttention for MI455X (gfx1250).
// Bandwidth-bound (~350MB traffic, ~6.4 GFLOP): fp32 WMMA
// (V_WMMA_F32_16X16X4_F32) keeps reference numerics; LDS stores the K dim
// pair-interleaved so every WMMA fragment is a single aligned ds_load_b64
// (no repack movs, conflict-free banks via 144-float2 pitch).
// ---------------------------------------------------------------------------

typedef __attribute__((ext_vector_type(2))) float v2f;
typedef __attribute__((ext_vector_type(8))) float v8f;

#define KC 32          // K-chunk staged in LDS per iteration (K total = 512)
#define PITCH 144      // LDS row pitch in float2 (288 dwords == 32 mod 64 -> no bank conflicts)

// ===========================================================================
// Kernel 1: per-n scores (3 fused GEMMs -> 2 accumulators) + softmax.
//   grid = 128 (one block per n), block = 512 threads = 16 waves (4x4 grid),
//   each wave owns a 32x32 output patch = 2x2 WMMA 16x16 tiles.
//   att_ws layout: [b][o][n][t]  (kernel 2 reads 256 contiguous floats/row)
// ===========================================================================
__global__ __launch_bounds__(512)
void ca_scores_softmax(const float* __restrict__ left,
                       const float* __restrict__ right,
                       const float* __restrict__ mid,
                       float* __restrict__ att_ws) {
  // 128KB LDS: pair-interleaved staging during GEMM, score planes afterwards.
  __shared__ v2f smem2[16384];
  v2f* sA0 = smem2;                    // [KC/2][PITCH]  {mid[b,c,2n],   mid[b,c+1,2n]}
  v2f* sA1 = smem2 + (KC / 2) * PITCH; //                {mid[b,c,2n+1], mid[b,c+1,2n+1]}
  v2f* sB0 = sA1 + (KC / 2) * PITCH;   //                {L[o,c,2n],     L[o,c+1,2n]}
  v2f* sB1 = sB0 + (KC / 2) * PITCH;   //                {L[o,c,2n+1],   L[o,c+1,2n+1]}

  const int n    = blockIdx.x;
  const int tid  = threadIdx.x;
  const int lane = tid & 31;
  const int wid  = tid >> 5;     // 0..15
  const int waveM = wid >> 2;    // 0..3
  const int waveN = wid & 3;     // 0..3

  const float* wsrc = (n < 64) ? left : right;          // concat(left,right) source
  const int wcol = (n < 64) ? (2 * n) : (2 * n - 128);
  const int mcol = 2 * n;

  v8f acc0[2][2];  // t=0 scores: S_b
  v8f acc1[2][2];  // t=1 scores: S_a + S_c
  for (int i = 0; i < 2; ++i)
    for (int j = 0; j < 2; ++j) { acc0[i][j] = {}; acc1[i][j] = {}; }

  const int mrow  = lane & 15;
  const int kpofs = lane >> 4;   // f32-WMMA A/B layout: lanes>=16 hold K+2,K+3

  for (int chunk = 0; chunk < 512 / KC; ++chunk) {
    const int c0 = chunk * KC;
    // ---- stage K-chunk into LDS, pair-interleaved over K ----
    for (int p = tid; p < (KC / 2) * 128; p += 512) {
      const int kp = p >> 7;    // 0..KC/2-1  (c-pair index)
      const int rr = p & 127;   // b (A) / o (B)
      const int cA = c0 + 2 * kp;
      const v2f m0 = *(const v2f*)(mid + ((size_t)rr * 512 + cA) * 256 + mcol);
      const v2f m1 = *(const v2f*)(mid + ((size_t)rr * 512 + cA + 1) * 256 + mcol);
      sA0[kp * PITCH + rr] = (v2f){m0.x, m1.x};
      sA1[kp * PITCH + rr] = (v2f){m0.y, m1.y};
      const v2f w0 = *(const v2f*)(wsrc + ((size_t)rr * 512 + cA) * 128 + wcol);
      const v2f w1 = *(const v2f*)(wsrc + ((size_t)rr * 512 + cA + 1) * 128 + wcol);
      sB0[kp * PITCH + rr] = (v2f){w0.x, w1.x};
      sB1[kp * PITCH + rr] = (v2f){w0.y, w1.y};
    }
    __syncthreads();

    // ---- WMMA over this chunk: fragments are single ds_load_b64 each ----
    for (int kk = 0; kk < KC; kk += 4) {
      const int kp = (kk >> 1) + kpofs;
      v2f a0[2], a1[2], b0[2], b1[2];
      for (int mi = 0; mi < 2; ++mi) {
        const int M = waveM * 32 + mi * 16 + mrow;
        a0[mi] = sA0[kp * PITCH + M];
        a1[mi] = sA1[kp * PITCH + M];
      }
      for (int ni = 0; ni < 2; ++ni) {
        const int N = waveN * 32 + ni * 16 + mrow;
        b0[ni] = sB0[kp * PITCH + N];
        b1[ni] = sB1[kp * PITCH + N];
      }
      for (int mi = 0; mi < 2; ++mi)
        for (int ni = 0; ni < 2; ++ni) {
          acc1[mi][ni] = __builtin_amdgcn_wmma_f32_16x16x4_f32(
              false, a0[mi], false, b0[ni], (short)0, acc1[mi][ni], false, false);
          acc1[mi][ni] = __builtin_amdgcn_wmma_f32_16x16x4_f32(
              false, a1[mi], false, b1[ni], (short)0, acc1[mi][ni], false, false);
          acc0[mi][ni] = __builtin_amdgcn_wmma_f32_16x16x4_f32(
              false, a0[mi], false, b1[ni], (short)0, acc0[mi][ni], false, false);
        }
    }
    __syncthreads();
  }

  // ---- dump scaled scores to LDS: sS[t][b][o], reusing staging memory ----
  float* sS = (float*)smem2;  // [2][128][128] = 32768 floats = 128KB
  {
    const int mofs = (lane >> 4) << 3;  // C/D layout: lanes>=16 hold M=r+8
    for (int mi = 0; mi < 2; ++mi)
      for (int ni = 0; ni < 2; ++ni) {
        const int N  = waveN * 32 + ni * 16 + mrow;
        const int Mb = waveM * 32 + mi * 16 + mofs;
        for (int r = 0; r < 8; ++r) {
          const int M = Mb + r;
          sS[M * 128 + N]         = 10.0f * acc0[mi][ni][r];  // t = 0
          sS[16384 + M * 128 + N] = 10.0f * acc1[mi][ni][r];  // t = 1
        }
      }
  }
  __syncthreads();

  // ---- stable softmax over o for each (b, t): 256 rows, one thread each ----
  if (tid < 256) {
    const int b  = tid & 127;
    const int tt = tid >> 7;
    const float* row = sS + tt * 16384 + b * 128;
    float mx = row[0];
    for (int o = 1; o < 128; ++o) mx = fmaxf(mx, row[o]);
    float sum = 0.0f;
    for (int o = 0; o < 128; ++o) sum += expf(row[o] - mx);
    const float inv = 1.0f / sum;
    // att_ws[((b*128 + o)*128 + n)*2 + tt]
    float* dst = att_ws + (size_t)b * 32768 + 2 * n + tt;
    for (int o = 0; o < 128; ++o)
      dst[(size_t)o * 256] = expf(row[o] - mx) * inv;
  }
}

// ===========================================================================
// Kernel 2: assemble both outputs. One wave per (b,c) row; lane L owns
// n = 4L..4L+3 -> 12 contiguous output floats -> 3 coalesced b128 stores
// per output tensor.
// ===========================================================================
__global__ __launch_bounds__(256)
void ca_assemble(const float* __restrict__ att_ws,
                 const float* __restrict__ sc00,
                 const float* __restrict__ sc10,
                 float* __restrict__ out) {
  const int lane = threadIdx.x & 31;
  const int w    = threadIdx.x >> 5;
  const int row  = blockIdx.x * 8 + w;   // row = b*512 + c, 0..65535
  const int c    = row & 511;
  const int b    = row >> 9;

  const size_t rbase = (size_t)row * 384;
  const float4 s0 = ((const float4*)(sc00 + rbase))[lane];  // centers out0: n=4L..4L+3
  const float4 s1 = ((const float4*)(sc10 + rbase))[lane];  // centers out1

  float4 a0 = make_float4(0.f, 0.f, 0.f, 0.f);  // (n0,t0)(n0,t1)(n1,t0)(n1,t1)
  float4 a1 = make_float4(0.f, 0.f, 0.f, 0.f);  // (n2,..)(n3,..)
  if (c < 128) {
    const float4* ap = (const float4*)(att_ws + ((size_t)b * 128 + c) * 256);
    a0 = ap[2 * lane];
    a1 = ap[2 * lane + 1];
  }

  const float4 o00 = make_float4(a0.x, a0.y, s0.x, a0.z);
  const float4 o01 = make_float4(a0.w, s0.y, a1.x, a1.y);
  const float4 o02 = make_float4(s0.z, a1.z, a1.w, s0.w);
  const float4 o10 = make_float4(a0.x, a0.y, s1.x, a0.z);
  const float4 o11 = make_float4(a0.w, s1.y, a1.x, a1.y);
  const float4 o12 = make_float4(s1.z, a1.z, a1.w, s1.w);

  float4* d0 = (float4*)(out + rbase) + lane * 3;
  float4* d1 = (float4*)(out + (size_t)128 * 512 * 384 + rbase) + lane * 3;
  d0[0] = o00; d0[1] = o01; d0[2] = o02;
  d1[0] = o10; d1[1] = o11; d1[2] = o12;
}

// ===========================================================================
extern "C" void kernel_launch(void* const* d_in, const int* in_sizes, int n_in,
                              void* d_out, int out_size, void* d_ws, size_t ws_size,
                              hipStream_t stream) {
  (void)in_sizes; (void)n_in; (void)out_size; (void)ws_size;
  const float* left  = (const float*)d_in[0];
  const float* right = (const float*)d_in[1];
  const float* mid   = (const float*)d_in[2];
  const float* sc00  = (const float*)d_in[3];
  // d_in[4] (sc01) and d_in[6] (sc11) are dead in the reference computation.
  const float* sc10  = (const float*)d_in[5];
  float* out    = (float*)d_out;
  float* att_ws = (float*)d_ws;   // needs 2*128*128*128*4 = 16 MB

  ca_scores_softmax<<<128, 512, 0, stream>>>(left, right, mid, att_ws);
  ca_assemble<<<8192, 256, 0, stream>>>(att_ws, sc00, sc10, out);
}